// InputNet_13228499271883
// MI455X (gfx1250) — compile-verified
//
#include <hip/hip_runtime.h>
#include <math.h>

// ---------------------------------------------------------------------------
// MI455X reasoning: this workload is ~2.5 MB in / ~1.26 MB out -> ~160 ns of
// HBM traffic at 23.3 TB/s. It is launch/latency bound with zero matmul, so
// WMMA does not apply; the CDNA5 paths that DO apply are wide b128 loads,
// global_prefetch_b8, and the async global->LDS gather (ASYNCcnt) for the
// scattered joint indices, which we use below via inline gfx1250 asm.
// ---------------------------------------------------------------------------

#define USE_ASYNC_GATHER 1

__device__ __constant__ short d_LIP[40] = {
    61, 146, 91, 181, 84, 17, 314, 405, 321, 375,
    291, 78, 95, 88, 178, 87, 14, 317, 402, 318,
    324, 308, 191, 80, 81, 82, 13, 312, 311, 310,
    415, 185, 40, 39, 37, 0, 267, 269, 270, 409};
__device__ __constant__ short d_SPOSE[8] = {500, 502, 504, 501, 503, 505, 512, 513};

__device__ __forceinline__ float nz(float v) { return __builtin_isnan(v) ? 0.0f : v; }

// ws layout: [0]=sum (double), [1]=sumsq (double), [2]=count of non-NaN (double)
__global__ void init_ws_kernel(double* ws) {
  ws[0] = 0.0; ws[1] = 0.0; ws[2] = 0.0;
}

__global__ __launch_bounds__(256)
void reduce_kernel(const float* __restrict__ x, int n, double* ws) {
  const int tid = threadIdx.x;
  const int gid = blockIdx.x * 256 + tid;
  const int n4  = n >> 2;

  double s = 0.0, s2 = 0.0, c = 0.0;
  if (gid < n4) {
    const float4* x4 = (const float4*)x;
    __builtin_prefetch(x4 + gid + 512, 0, 0);   // -> global_prefetch_b8
    float4 v = x4[gid];
    float e0 = v.x, e1 = v.y, e2 = v.z, e3 = v.w;
    if (!__builtin_isnan(e0)) { s += (double)e0; s2 += (double)e0 * (double)e0; c += 1.0; }
    if (!__builtin_isnan(e1)) { s += (double)e1; s2 += (double)e1 * (double)e1; c += 1.0; }
    if (!__builtin_isnan(e2)) { s += (double)e2; s2 += (double)e2 * (double)e2; c += 1.0; }
    if (!__builtin_isnan(e3)) { s += (double)e3; s2 += (double)e3 * (double)e3; c += 1.0; }
  }
  // tail (n not multiple of 4) handled by one thread
  if (blockIdx.x == 0 && tid == 0) {
    for (int i = (n4 << 2); i < n; ++i) {
      float f = x[i];
      if (!__builtin_isnan(f)) { s += (double)f; s2 += (double)f * (double)f; c += 1.0; }
    }
  }

  __shared__ double sh[256];
  // sum
  sh[tid] = s; __syncthreads();
  for (int o = 128; o > 0; o >>= 1) { if (tid < o) sh[tid] += sh[tid + o]; __syncthreads(); }
  if (tid == 0) atomicAdd(&ws[0], sh[0]);       // global_atomic_add_f64
  __syncthreads();
  // sumsq
  sh[tid] = s2; __syncthreads();
  for (int o = 128; o > 0; o >>= 1) { if (tid < o) sh[tid] += sh[tid + o]; __syncthreads(); }
  if (tid == 0) atomicAdd(&ws[1], sh[0]);
  __syncthreads();
  // count
  sh[tid] = c; __syncthreads();
  for (int o = 128; o > 0; o >>= 1) { if (tid < o) sh[tid] += sh[tid + o]; __syncthreads(); }
  if (tid == 0) atomicAdd(&ws[2], sh[0]);
}

// One block per output frame. Dynamic LDS P[810]:
//   P[f*270 + p*3 + c], f=0 prev frame, f=1 current, f=2 next; 90 joints x 3.
__global__ __launch_bounds__(256)
void gather_kernel(const float* __restrict__ xyz, const double* __restrict__ ws,
                   float* __restrict__ out, int start, int Lout)
{
  extern __shared__ float P[];
  const int t   = blockIdx.x;
  const int tid = threadIdx.x;
  const int src = start + t;
  const int lastFrame = start + Lout - 1;

  const double sum = ws[0], sumsq = ws[1], cnt = ws[2];
  const float  m  = (float)(sum / cnt);
  const float  sd = (float)sqrt((sumsq - sum * sum / cnt) / (cnt - 1.0));

  // ---- scattered gather of 810 scalars: async global -> LDS (ASYNCcnt path)
  for (int idx = tid; idx < 810; idx += 256) {
    int f = idx / 270;
    int r = idx - f * 270;
    int p = r / 3;
    int c = r - p * 3;
    int frame = src + f - 1;                       // prev/cur/next
    frame = frame < start ? start : (frame > lastFrame ? lastFrame : frame);
    int joint;
    if      (p < 21) joint = 468 + p;              // lhand
    else if (p < 42) joint = 501 + p;              // rhand: 522 + (p-21)
    else if (p < 82) joint = d_LIP[p - 42];
    else             joint = d_SPOSE[p - 82];
    unsigned goff = (unsigned)(((frame * 543 + joint) * 3 + c) * 4);
#if USE_ASYNC_GATHER
    unsigned loff = (unsigned)(idx * 4);           // dynamic LDS starts at offset 0
    asm volatile("global_load_async_to_lds_b32 %0, %1, %2"
                 :: "v"(loff), "v"(goff),
                    "s"((unsigned long long)(size_t)xyz)
                 : "memory");
#else
    P[idx] = xyz[goff >> 2];
#endif
  }
#if USE_ASYNC_GATHER
  asm volatile("s_wait_asynccnt 0" ::: "memory");
#endif
  __syncthreads();

  // ---- normalize in place: (x - m) / s
  for (int idx = tid; idx < 810; idx += 256)
    P[idx] = (P[idx] - m) / sd;
  __syncthreads();

  // ---- emit row: [pts(270) | df(270) | db(270) | ld(210) | rd(210)]
  float* orow = out + (size_t)t * 1230;
  const float* Pp = P;         // prev
  const float* Pc = P + 270;   // cur
  const float* Pn = P + 540;   // next
  for (int w = tid; w < 690; w += 256) {
    if (w < 270) {
      float a = Pc[w];
      orow[w]       = nz(a);
      orow[270 + w] = (t < Lout - 1) ? nz(a - Pn[w]) : 0.0f;  // pts[t]-pts[t+1]
      orow[540 + w] = (t > 0)        ? nz(a - Pp[w]) : 0.0f;  // pts[t]-pts[t-1]
    } else {
      int k    = w - 270;                 // 0..419
      int hand = (k >= 210) ? 1 : 0;      // 0 = lhand, 1 = rhand
      int kk   = k - (hand ? 210 : 0);
      // decode strict-upper-triangle pair (i,j), 0<=i<j<21, row-major
      int i = 0, rem = kk;
      while (rem >= 20 - i) { rem -= 20 - i; ++i; }
      int j = i + 1 + rem;
      int off = hand * 21;
      const float* A = Pc + (off + i) * 3;
      const float* B = Pc + (off + j) * 3;
      float dx = A[0] - B[0];
      float dy = A[1] - B[1];
      orow[810 + k] = nz(sqrtf(dx * dx + dy * dy));
    }
  }
}

extern "C" void kernel_launch(void* const* d_in, const int* in_sizes, int n_in,
                              void* d_out, int out_size, void* d_ws, size_t ws_size,
                              hipStream_t stream) {
  const float* xyz = (const float*)d_in[0];
  float* out = (float*)d_out;
  double* ws = (double*)d_ws;

  const int n    = in_sizes[0];         // 384*543*3 = 625536
  const int L    = n / (543 * 3);       // 384
  const int Lout = L > 256 ? 256 : L;   // 256
  const int start = L > 256 ? (L - 256) / 2 : 0;  // 64

  init_ws_kernel<<<1, 1, 0, stream>>>(ws);

  const int n4 = n >> 2;
  int rb = (n4 + 255) / 256; if (rb < 1) rb = 1;
  reduce_kernel<<<rb, 256, 0, stream>>>(xyz, n, ws);

  gather_kernel<<<Lout, 256, 810 * sizeof(float), stream>>>(xyz, ws, out, start, Lout);
}